// DigitCaps_21766894256556
// MI455X (gfx1250) — compile-verified
//
#include <hip/hip_runtime.h>
#include <math.h>

// Problem constants (from reference setup_inputs)
#define B_    256
#define R_    2304
#define C_    10
#define O_    16
#define I_    8
#define K_RI  (R_ * I_)   // 18432
#define N_CO  (C_ * O_)   // 160
#define SPLITK 16         // s-GEMM split-K: K chunk = 1152 -> 288 WMMA steps/wave, 2560 waves

typedef float v2f __attribute__((ext_vector_type(2)));
typedef float v8f __attribute__((ext_vector_type(8)));

// ---------------------------------------------------------------- utilities
__global__ void zero_kernel(float* __restrict__ p, int n) {
    int t = blockIdx.x * blockDim.x + threadIdx.x;
    if (t < n) p[t] = 0.0f;
}

// Xt[k, b] = X[b, k]  (one-time transpose so gemm_t A-fragments are contiguous)
__global__ void transpose_x_kernel(const float* __restrict__ X, float* __restrict__ Xt) {
    int t = blockIdx.x * blockDim.x + threadIdx.x;      // t = k*B_ + b (coalesced write)
    if (t >= B_ * K_RI) return;
    const int k = t / B_, b = t - k * B_;
    Xt[t] = X[(size_t)b * K_RI + k];
}

// softmax over axis 0 (R) for each column c of bij [R, C]
__global__ void softmax_kernel(const float* __restrict__ bij, float* __restrict__ cij) {
    __shared__ float red[256];
    const int c   = blockIdx.x;      // 0..C_-1
    const int tid = threadIdx.x;     // R_/256 == 9 exactly
    float m = -3.402823466e38f;
    for (int r = tid; r < R_; r += 256) m = fmaxf(m, bij[r * C_ + c]);
    red[tid] = m; __syncthreads();
    for (int s = 128; s > 0; s >>= 1) {
        if (tid < s) red[tid] = fmaxf(red[tid], red[tid + s]);
        __syncthreads();
    }
    m = red[0]; __syncthreads();
    float sum = 0.0f;
    for (int r = tid; r < R_; r += 256) sum += __expf(bij[r * C_ + c] - m);
    red[tid] = sum; __syncthreads();
    for (int s = 128; s > 0; s >>= 1) {
        if (tid < s) red[tid] += red[tid + s];
        __syncthreads();
    }
    const float inv = 1.0f / red[0];
    for (int r = tid; r < R_; r += 256)
        cij[r * C_ + c] = __expf(bij[r * C_ + c] - m) * inv;
}

// WcT[(c*O+o)*K_RI + r*I+i] = cij[r,c] * W[r,c,o,i]   (reweight into n-major layout,
// so B-fragment K-pairs are contiguous float2 in the s-GEMM)
__global__ void scale_w_kernel(const float* __restrict__ W, const float* __restrict__ cij,
                               float* __restrict__ WcT) {
    int t = blockIdx.x * blockDim.x + threadIdx.x;      // t = n*K_RI + k (coalesced write)
    if (t >= K_RI * N_CO) return;
    const int n = t / K_RI, k = t - n * K_RI;
    const int c = n / O_,   o = n - c * O_;
    const int r = k / I_,   i = k - r * I_;
    WcT[t] = cij[r * C_ + c] * W[(((r * C_ + c) * O_) + o) * I_ + i];
}

// ---------------------------------------------------------------- fp32 WMMA GEMMs
// Fragment layout per ISA 7.12.2 (V_WMMA_F32_16X16X4_F32, wave32):
//   A (16x4): lanes 0-15 -> M=lane, K={0,1}; lanes 16-31 -> M=lane-16, K={2,3}
//   B (4x16): lanes 0-15 -> N=lane, K={0,1}; lanes 16-31 -> N=lane-16, K={2,3}
//   D (16x16): VGPR v -> row v (lanes 0-15) / row 8+v (lanes 16-31)
// Both operands are laid out reduction-major so each fragment is ONE global_load_b64.

// s[b, co] = sum_k X[b, k] * WcT[co, k];  M=256, N=160, K=18432, split-K + atomics.
__global__ void gemm_s_kernel(const float* __restrict__ X,    // [B_, K_RI]
                              const float* __restrict__ WcT,  // [N_CO, K_RI]
                              float* __restrict__ S) {        // [B_, N_CO], pre-zeroed
    const int lane = threadIdx.x;
    const int half = lane >> 4, ln = lane & 15;
    const int m0 = blockIdx.x * 16, n0 = blockIdx.y * 16;
    const float* arow = X   + (size_t)(m0 + ln) * K_RI + 2 * half;
    const float* brow = WcT + (size_t)(n0 + ln) * K_RI + 2 * half;
    const int k0 = blockIdx.z * (K_RI / SPLITK);
    const int k1 = k0 + (K_RI / SPLITK);
    v8f acc = {};
#pragma unroll 8
    for (int k = k0; k < k1; k += 4) {
        v2f a = *(const v2f*)(arow + k);
        v2f b = *(const v2f*)(brow + k);
        acc = __builtin_amdgcn_wmma_f32_16x16x4_f32(false, a, false, b,
                                                    (short)0, acc, false, false);
    }
#pragma unroll
    for (int vr = 0; vr < 8; ++vr)
        atomicAdd(&S[(size_t)(m0 + vr + 8 * half) * N_CO + n0 + ln], acc[vr]);
}

// T[k, co] = sum_b Xt[k, b] * Vt[co, b];  M=18432, N=160, K=256.
__global__ void gemm_t_kernel(const float* __restrict__ Xt,  // [K_RI, B_]
                              const float* __restrict__ Vt,  // [N_CO, B_]
                              float* __restrict__ T) {       // [K_RI, N_CO]
    const int lane = threadIdx.x;
    const int half = lane >> 4, ln = lane & 15;
    const int m0 = blockIdx.x * 16, n0 = blockIdx.y * 16;
    const float* arow = Xt + (size_t)(m0 + ln) * B_ + 2 * half;
    const float* brow = Vt + (size_t)(n0 + ln) * B_ + 2 * half;
    v8f acc = {};
#pragma unroll 8
    for (int k = 0; k < B_; k += 4) {
        v2f a = *(const v2f*)(arow + k);
        v2f b = *(const v2f*)(brow + k);
        acc = __builtin_amdgcn_wmma_f32_16x16x4_f32(false, a, false, b,
                                                    (short)0, acc, false, false);
    }
#pragma unroll
    for (int vr = 0; vr < 8; ++vr)
        T[(size_t)(m0 + vr + 8 * half) * N_CO + n0 + ln] = acc[vr];
}

// bij[r,c] += (1/B) * sum_{o,i} W[r,c,o,i] * T[r*I+i, c*O+o]
__global__ void update_b_kernel(const float* __restrict__ W, const float* __restrict__ T,
                                float* __restrict__ bij) {
    int t = blockIdx.x * blockDim.x + threadIdx.x;
    if (t >= R_ * C_) return;
    const int r = t / C_, c = t - r * C_;
    float acc = 0.0f;
    for (int o = 0; o < O_; ++o)
        for (int i = 0; i < I_; ++i)
            acc += W[((r * C_ + c) * O_ + o) * I_ + i] * T[(size_t)(r * I_ + i) * N_CO + c * O_ + o];
    bij[t] += acc * (1.0f / (float)B_);
}

// v = squash(s) along O for each (b, c):  v = sq*s / ((1+sq)*sqrt(sq))
// transposed==0: write v[b*N_CO + c*O + o]   (final output layout [B,C,O])
// transposed==1: write v[(c*O+o)*B_ + b]     (n-major, feeds gemm_t B operand)
__global__ void squash_kernel(const float* __restrict__ S, float* __restrict__ Vout,
                              int transposed) {
    int t = blockIdx.x * blockDim.x + threadIdx.x;
    if (t >= B_ * C_) return;
    const int b = t / C_, c = t - b * C_;
    const float* s = S + (size_t)b * N_CO + c * O_;
    float sq = 0.0f;
    for (int o = 0; o < O_; ++o) sq += s[o] * s[o];
    const float sc = sq / ((1.0f + sq) * sqrtf(sq));
    if (transposed) {
        for (int o = 0; o < O_; ++o) Vout[(size_t)(c * O_ + o) * B_ + b] = s[o] * sc;
    } else {
        for (int o = 0; o < O_; ++o) Vout[(size_t)b * N_CO + c * O_ + o] = s[o] * sc;
    }
}

// ---------------------------------------------------------------- driver
extern "C" void kernel_launch(void* const* d_in, const int* in_sizes, int n_in,
                              void* d_out, int out_size, void* d_ws, size_t ws_size,
                              hipStream_t stream) {
    const float* x = (const float*)d_in[0];   // [B, R, I] == [B, K_RI] row-major
    const float* W = (const float*)d_in[1];   // [R, C, O, I]
    float* out = (float*)d_out;               // [B, C, O, 1] -> flat B*C*O

    float* ws  = (float*)d_ws;                // ~43 MB total
    float* bij = ws;                          // R*C       = 23,040
    float* cij = bij + R_ * C_;               // 23,040
    float* WcT = cij + R_ * C_;               // N_CO*K_RI = 2,949,120
    float* T   = WcT + (size_t)K_RI * N_CO;   // 2,949,120
    float* Xt  = T   + (size_t)K_RI * N_CO;   // K_RI*B_   = 4,718,592
    float* S   = Xt  + (size_t)K_RI * B_;     // B*N_CO    = 40,960
    float* Vt  = S   + (size_t)B_ * N_CO;     // 40,960

    zero_kernel<<<(R_ * C_ + 255) / 256, 256, 0, stream>>>(bij, R_ * C_);
    transpose_x_kernel<<<(B_ * K_RI + 255) / 256, 256, 0, stream>>>(x, Xt);

    for (int it = 0; it < 3; ++it) {
        softmax_kernel<<<C_, 256, 0, stream>>>(bij, cij);
        scale_w_kernel<<<(K_RI * N_CO + 255) / 256, 256, 0, stream>>>(W, cij, WcT);
        zero_kernel<<<(B_ * N_CO + 255) / 256, 256, 0, stream>>>(S, B_ * N_CO);
        gemm_s_kernel<<<dim3(B_ / 16, N_CO / 16, SPLITK), 32, 0, stream>>>(x, WcT, S);
        if (it == 2) {
            squash_kernel<<<(B_ * C_ + 255) / 256, 256, 0, stream>>>(S, out, 0);
        } else {
            squash_kernel<<<(B_ * C_ + 255) / 256, 256, 0, stream>>>(S, Vt, 1);
            gemm_t_kernel<<<dim3(K_RI / 16, N_CO / 16), 32, 0, stream>>>(Xt, Vt, T);
            update_b_kernel<<<(R_ * C_ + 255) / 256, 256, 0, stream>>>(W, T, bij);
        }
    }
}